// ODEBlock_84524956385767
// MI455X (gfx1250) — compile-verified
//
#include <hip/hip_runtime.h>

typedef float f32x2 __attribute__((ext_vector_type(2)));
typedef float f32x4 __attribute__((ext_vector_type(4)));
typedef float f32x8 __attribute__((ext_vector_type(8)));

// ---------------------------------------------------------------------------
// Fused GEMM:  C[M,N] = (A[M,K] @ B[K,N]) + bias[N], optional tanh epilogue.
// fp32 WMMA path: v_wmma_f32_16x16x4_f32 (wave32, CDNA5).
//   Block: 128 threads = 4 waves, 64x128 output tile, BK = 32.
//   Wave (wm,wn in 2x2) owns a 32x64 sub-tile = 2x4 WMMA accumulators.
//   Both As and Bs are stored K-contiguous (Bs transposed) so every WMMA
//   fragment is a single 8-byte LDS load -> ds_load_2addr_b64, no pack movs.
// ---------------------------------------------------------------------------
template <bool FUSE_TANH>
__global__ __launch_bounds__(128) void wmma_gemm_f32(
    const float* __restrict__ A,    // M x K row-major
    const float* __restrict__ B,    // K x N row-major
    const float* __restrict__ bias, // N
    float* __restrict__ C,          // M x N row-major
    int M, int N, int K) {
  constexpr int BM = 64, BN = 128, BK = 32;
  constexpr int LDK = BK + 2;                // 34-float stride: bank spread, keeps b64 align
  __shared__ float As[BM][LDK];              // row-major (K contiguous)
  __shared__ float Bs[BN][LDK];              // transposed  (K contiguous)

  const int tid  = threadIdx.x;
  const int lane = tid & 31;
  const int wave = tid >> 5;
  const int wm   = wave >> 1;  // 0..1 (M direction)
  const int wn   = wave & 1;   // 0..1 (N direction)
  const int l16  = lane & 15;
  const int hi   = lane >> 4;  // 0 or 1

  const int rowBase = blockIdx.y * BM;
  const int colBase = blockIdx.x * BN;

  f32x8 acc[2][4] = {};

  // staging index helpers
  const int atr = tid >> 3;  // 0..15  A: row within pass (16 rows/pass, 4 passes)
  const int atc = tid & 7;   // 0..7   A: float4 column
  const int btr = tid >> 5;  // 0..3   B: k-row within pass (4 rows/pass, 8 passes)
  const int btc = tid & 31;  // 0..31  B: float4 column

  for (int k0 = 0; k0 < K; k0 += BK) {
    // ---- stage A tile 64x32 ----
#pragma unroll
    for (int p = 0; p < 4; ++p) {
      const int r = atr + p * 16;
      f32x4 v = *(const f32x4*)(A + (size_t)(rowBase + r) * K + k0 + atc * 4);
      As[r][atc * 4 + 0] = v[0];
      As[r][atc * 4 + 1] = v[1];
      As[r][atc * 4 + 2] = v[2];
      As[r][atc * 4 + 3] = v[3];
    }
    // ---- stage B tile 32x128, transposed into Bs[n][k] ----
#pragma unroll
    for (int p = 0; p < 8; ++p) {
      const int r = btr + p * 4;  // k index within tile
      f32x4 v = *(const f32x4*)(B + (size_t)(k0 + r) * N + colBase + btc * 4);
      Bs[btc * 4 + 0][r] = v[0];
      Bs[btc * 4 + 1][r] = v[1];
      Bs[btc * 4 + 2][r] = v[2];
      Bs[btc * 4 + 3][r] = v[3];
    }
    // prefetch next K tile into cache (global_prefetch_b8); uniform branch
    if (k0 + BK < K) {
      __builtin_prefetch(A + (size_t)(rowBase + atr) * K + (k0 + BK) + atc * 4, 0, 0);
      __builtin_prefetch(B + (size_t)(k0 + BK + btr) * N + colBase + btc * 4, 0, 0);
    }
    __syncthreads();

#pragma unroll
    for (int kk = 0; kk < BK; kk += 4) {
      // A fragment (16x4): lane l16 = M row, VGPR pair = K {kk+2*hi, kk+2*hi+1}
      f32x2 af[2];
#pragma unroll
      for (int tm = 0; tm < 2; ++tm)
        af[tm] = *(const f32x2*)&As[wm * 32 + tm * 16 + l16][kk + 2 * hi];
      // B fragment (4x16): lane l16 = N col, VGPR pair = K {kk+2*hi, kk+2*hi+1}
      f32x2 bf[4];
#pragma unroll
      for (int tn = 0; tn < 4; ++tn)
        bf[tn] = *(const f32x2*)&Bs[wn * 64 + tn * 16 + l16][kk + 2 * hi];
#pragma unroll
      for (int tm = 0; tm < 2; ++tm)
#pragma unroll
        for (int tn = 0; tn < 4; ++tn)
          acc[tm][tn] = __builtin_amdgcn_wmma_f32_16x16x4_f32(
              false, af[tm], false, bf[tn], (short)0, acc[tm][tn], false, false);
    }
    __syncthreads();
  }

  // ---- epilogue: bias (+ tanh); C/D layout: VGPR i -> row i + 8*hi, col l16 ----
#pragma unroll
  for (int tm = 0; tm < 2; ++tm) {
#pragma unroll
    for (int tn = 0; tn < 4; ++tn) {
      const int col = colBase + wn * 64 + tn * 16 + l16;
      const float bv = bias[col];
#pragma unroll
      for (int i = 0; i < 8; ++i) {
        const int row = rowBase + wm * 32 + tm * 16 + i + 8 * hi;
        float v = acc[tm][tn][i] + bv;
        if (FUSE_TANH) v = tanhf(v);
        C[(size_t)row * N + col] = v;
      }
    }
  }
}

// ---------------------------------------------------------------------------
// out = y + h * (c1*k1 + ... + c6*k6), float4-vectorized.
// Uniform scalar branches skip zero-coefficient buffers (never touches
// uninitialized memory; EXEC stays all-ones).
// ---------------------------------------------------------------------------
__global__ __launch_bounds__(256) void ode_axpy_kernel(
    const float* __restrict__ y,
    const float* __restrict__ k1, const float* __restrict__ k2,
    const float* __restrict__ k3, const float* __restrict__ k4,
    const float* __restrict__ k5, const float* __restrict__ k6,
    float c1, float c2, float c3, float c4, float c5, float c6,
    float h, float* __restrict__ out, int n4) {
  const int i = blockIdx.x * blockDim.x + threadIdx.x;
  if (i >= n4) return;
  f32x4 acc = ((const f32x4*)k1)[i] * c1;
  if (c2 != 0.0f) acc += ((const f32x4*)k2)[i] * c2;
  if (c3 != 0.0f) acc += ((const f32x4*)k3)[i] * c3;
  if (c4 != 0.0f) acc += ((const f32x4*)k4)[i] * c4;
  if (c5 != 0.0f) acc += ((const f32x4*)k5)[i] * c5;
  if (c6 != 0.0f) acc += ((const f32x4*)k6)[i] * c6;
  ((f32x4*)out)[i] = ((const f32x4*)y)[i] + acc * h;
}

__global__ __launch_bounds__(256) void copy4_kernel(const f32x4* __restrict__ in,
                                                    f32x4* __restrict__ out, int n4) {
  const int i = blockIdx.x * blockDim.x + threadIdx.x;
  if (i < n4) out[i] = in[i];
}

// ---------------------------------------------------------------------------
extern "C" void kernel_launch(void* const* d_in, const int* in_sizes, int n_in,
                              void* d_out, int out_size, void* d_ws, size_t ws_size,
                              hipStream_t stream) {
  (void)in_sizes; (void)n_in; (void)out_size; (void)ws_size;
  const float* x  = (const float*)d_in[0];
  const float* W1 = (const float*)d_in[1];
  const float* b1 = (const float*)d_in[2];
  const float* W2 = (const float*)d_in[3];
  const float* b2 = (const float*)d_in[4];
  float* out = (float*)d_out;

  constexpr int M = 512, D = 1024, H = 2048;
  constexpr int YN = M * D;  // 524288 elements (2 MB)
  float* ws = (float*)d_ws;
  float* y  = ws;                          // [0, YN)
  float* kb[6];
  for (int i = 0; i < 6; ++i) kb[i] = ws + (size_t)(1 + i) * YN;  // k1..k6
  float* yt = ws + (size_t)7 * YN;         // stage input
  float* z  = ws + (size_t)8 * YN;         // hidden 512x2048 (2*YN)

  const float h = 1.0f / 16.0f;
  const int n4 = YN / 4;
  const dim3 eb(256), eg(n4 / 256);
  const dim3 g1(H / 128, M / 64), g2(D / 128, M / 64), tb(128);

  auto feval = [&](const float* yin, float* kout) {
    wmma_gemm_f32<true ><<<g1, tb, 0, stream>>>(yin, W1, b1, z,    M, H, D);
    wmma_gemm_f32<false><<<g2, tb, 0, stream>>>(z,   W2, b2, kout, M, D, H);
  };
  auto axpy = [&](float c1, float c2, float c3, float c4, float c5, float c6,
                  float* o) {
    ode_axpy_kernel<<<eg, eb, 0, stream>>>(y, kb[0], kb[1], kb[2], kb[3], kb[4],
                                           kb[5], c1, c2, c3, c4, c5, c6, h, o, n4);
  };

  copy4_kernel<<<eg, eb, 0, stream>>>((const f32x4*)x, (f32x4*)y, n4);

  for (int s = 0; s < 16; ++s) {
    feval(y, kb[0]);
    axpy(0.2f, 0.f, 0.f, 0.f, 0.f, 0.f, yt);
    feval(yt, kb[1]);
    axpy(3.0f / 40.0f, 9.0f / 40.0f, 0.f, 0.f, 0.f, 0.f, yt);
    feval(yt, kb[2]);
    axpy(44.0f / 45.0f, -56.0f / 15.0f, 32.0f / 9.0f, 0.f, 0.f, 0.f, yt);
    feval(yt, kb[3]);
    axpy(19372.0f / 6561.0f, -25360.0f / 2187.0f, 64448.0f / 6561.0f,
         -212.0f / 729.0f, 0.f, 0.f, yt);
    feval(yt, kb[4]);
    axpy(9017.0f / 3168.0f, -355.0f / 33.0f, 46732.0f / 5247.0f, 49.0f / 176.0f,
         -5103.0f / 18656.0f, 0.f, yt);
    feval(yt, kb[5]);
    float* tgt = (s == 15) ? out : y;
    axpy(35.0f / 384.0f, 0.f, 500.0f / 1113.0f, 125.0f / 192.0f,
         -2187.0f / 6784.0f, 11.0f / 84.0f, tgt);
  }
}